// PPGML_TripletLoss_82317343195174
// MI455X (gfx1250) — compile-verified
//
#include <hip/hip_runtime.h>
#include <hip/hip_bf16.h>
#include <math.h>

// ---------------------------------------------------------------------------
// PPGML_TripletLoss for MI455X (gfx1250, wave32).
// N=384, D=512, DIMS={4,5,6}, 9 classes, 5 neighbors.
// ---------------------------------------------------------------------------

#define NPTS   384
#define DFEAT  512
#define NN2    (NPTS * NPTS)   // 147456
#define KSUB   8               // subspace width (>= max(DIMS)=6)

typedef float v2f __attribute__((ext_vector_type(2)));
typedef float v8f __attribute__((ext_vector_type(8)));

// ---------------------------------------------------------------------------
// Gram = X X^T with f32 WMMA (V_WMMA_F32_16X16X4_F32), one wave per 16x16 tile.
// A layout (16x4, MxK): lanes 0-15 -> M=lane, VGPR{0,1}=K{0,1}; lanes 16-31 -> K{2,3}.
// B layout (4x16, KxN): mirrored with N in lane position. C/D: 8 VGPRs per lane.
// ---------------------------------------------------------------------------
__global__ __launch_bounds__(32) void ppgml_gram_wmma(const float* __restrict__ X,
                                                      float* __restrict__ G) {
  const int tile = blockIdx.x;            // 24*24 tiles
  const int tm = (tile / 24) * 16;
  const int tn = (tile % 24) * 16;
  const int lane = threadIdx.x;
  const int r    = lane & 15;
  const int koff = (lane >> 4) * 2;
  const float* __restrict__ arow = X + (tm + r) * DFEAT + koff;
  const float* __restrict__ brow = X + (tn + r) * DFEAT + koff;
  v8f acc = {};
  for (int k = 0; k < DFEAT; k += 4) {
    v2f a, b;
    a.x = arow[k]; a.y = arow[k + 1];
    b.x = brow[k]; b.y = brow[k + 1];
    acc = __builtin_amdgcn_wmma_f32_16x16x4_f32(false, a, false, b,
                                                (short)0, acc, false, false);
  }
  const int col   = lane & 15;
  const int rbase = (lane >> 4) * 8;
#pragma unroll
  for (int v = 0; v < 8; ++v)
    G[(tm + rbase + v) * NPTS + tn + col] = acc[v];
}

// d[i][j] = sqrt(max(sq_i + sq_j - 2*G[i][j], 0))
__global__ void ppgml_dist(const float* __restrict__ G, float* __restrict__ d) {
  int t = blockIdx.x * blockDim.x + threadIdx.x;
  if (t >= NN2) return;
  int i = t / NPTS, j = t - i * NPTS;
  float d2 = G[i * NPTS + i] + G[j * NPTS + j] - 2.0f * G[t];
  d[t] = sqrtf(fmaxf(d2, 0.0f));
}

// 6 smallest per row (includes self at d=0), one thread per row
__global__ void ppgml_knn(const float* __restrict__ d, int* __restrict__ idx) {
  int i = blockIdx.x * blockDim.x + threadIdx.x;
  if (i >= NPTS) return;
  float bd[6]; int bi[6];
#pragma unroll
  for (int t = 0; t < 6; ++t) { bd[t] = INFINITY; bi[t] = 0; }
  for (int j = 0; j < NPTS; ++j) {
    float v = d[i * NPTS + j];
    if (v < bd[5]) {
      int t = 5;
      while (t > 0 && v < bd[t - 1]) { bd[t] = bd[t - 1]; bi[t] = bi[t - 1]; --t; }
      bd[t] = v; bi[t] = j;
    }
  }
#pragma unroll
  for (int t = 0; t < 6; ++t) idx[i * 6 + t] = bi[t];
}

__global__ void ppgml_ginit(float* __restrict__ G) {
  int t = blockIdx.x * blockDim.x + threadIdx.x;
  if (t < NN2) G[t] = INFINITY;
}

__global__ void ppgml_gscatter(const float* __restrict__ d, const int* __restrict__ idx,
                               float* __restrict__ G) {
  int t = blockIdx.x * blockDim.x + threadIdx.x;
  if (t >= NPTS * 6) return;
  int i = t / 6;
  int j = idx[t];
  G[i * NPTS + j] = d[i * NPTS + j];
}

__global__ void ppgml_gsym(float* __restrict__ G) {
  int t = blockIdx.x * blockDim.x + threadIdx.x;
  if (t >= NN2) return;
  int i = t / NPTS, j = t - i * NPTS;
  float a = G[i * NPTS + j], b = G[j * NPTS + i];
  G[i * NPTS + j] = fminf(a, b);  // idempotent under races
}

// ---------------------------------------------------------------------------
// Floyd-Warshall: single persistent workgroup; 384 dependent steps.
// Row k / col k are invariant within step k (diag is 0), so in-place is safe.
// ---------------------------------------------------------------------------
__global__ __launch_bounds__(1024) void ppgml_floyd(float* __restrict__ Dm) {
  __shared__ float rowk[NPTS];
  __shared__ float colk[NPTS];
  const int tid = threadIdx.x;
  for (int k = 0; k < NPTS; ++k) {
    if (tid < NPTS) {
      rowk[tid] = Dm[k * NPTS + tid];
      colk[tid] = Dm[tid * NPTS + k];
      if (k + 1 < NPTS)
        __builtin_prefetch(&Dm[(k + 1) * NPTS + tid], 0, 3);  // global_prefetch
    }
    __syncthreads();
    for (int e = tid; e < NN2; e += 1024) {
      int i = e / NPTS, j = e - i * NPTS;
      float cand = colk[i] + rowk[j];
      if (cand < Dm[e]) Dm[e] = cand;
    }
    __threadfence_block();
    __syncthreads();
  }
}

// max over finite entries (values >= 0 -> uint-bits compare is monotonic)
__global__ void ppgml_fmax(const float* __restrict__ Dm, unsigned* __restrict__ fb) {
  __shared__ float red[256];
  int t = blockIdx.x * blockDim.x + threadIdx.x;
  float loc = 0.0f;
  for (int e = t; e < NN2; e += gridDim.x * blockDim.x) {
    float v = Dm[e];
    if (isfinite(v)) loc = fmaxf(loc, v);
  }
  red[threadIdx.x] = loc; __syncthreads();
  for (int s = 128; s > 0; s >>= 1) {
    if (threadIdx.x < s) red[threadIdx.x] = fmaxf(red[threadIdx.x], red[threadIdx.x + s]);
    __syncthreads();
  }
  if (threadIdx.x == 0) atomicMax(fb, __float_as_uint(red[0]));
}

// B = -0.5*Dg^2 (inf replaced by fmax), per-row mean + grand total
__global__ __launch_bounds__(128) void ppgml_bmat(float* __restrict__ Dm,
                                                  float* __restrict__ rm,
                                                  float* __restrict__ total,
                                                  const unsigned* __restrict__ fb) {
  __shared__ float red[128];
  const int i = blockIdx.x;
  const float fmx = __uint_as_float(*fb);
  float s = 0.0f;
  for (int j = threadIdx.x; j < NPTS; j += 128) {
    float v = Dm[i * NPTS + j];
    if (!isfinite(v)) v = fmx;
    float b = -0.5f * v * v;
    Dm[i * NPTS + j] = b;
    s += b;
  }
  red[threadIdx.x] = s; __syncthreads();
  for (int st = 64; st > 0; st >>= 1) {
    if (threadIdx.x < st) red[threadIdx.x] += red[threadIdx.x + st];
    __syncthreads();
  }
  if (threadIdx.x == 0) { rm[i] = red[0] / (float)NPTS; atomicAdd(total, red[0]); }
}

__global__ void ppgml_center(float* __restrict__ K, const float* __restrict__ rm,
                             const float* __restrict__ total) {
  int t = blockIdx.x * blockDim.x + threadIdx.x;
  if (t >= NN2) return;
  int i = t / NPTS, j = t - i * NPTS;
  float mean = *total / (float)NN2;
  K[t] = K[t] - rm[i] - rm[j] + mean;
}

__global__ void ppgml_sumall(const float* __restrict__ x, int n, float* __restrict__ out,
                             int squared) {
  __shared__ float red[256];
  int t = blockIdx.x * blockDim.x + threadIdx.x;
  float s = 0.0f;
  for (int i = t; i < n; i += gridDim.x * blockDim.x) {
    float v = x[i];
    s += squared ? v * v : v;
  }
  red[threadIdx.x] = s; __syncthreads();
  for (int st = 128; st > 0; st >>= 1) {
    if (threadIdx.x < st) red[threadIdx.x] += red[threadIdx.x + st];
    __syncthreads();
  }
  if (threadIdx.x == 0) atomicAdd(out, red[0]);
}

// ---------------------------------------------------------------------------
// Shifted subspace iteration for top eigenpairs of K (column-major V[q][i]).
// ---------------------------------------------------------------------------
__global__ void ppgml_initv(float* __restrict__ V) {
  int t = blockIdx.x * blockDim.x + threadIdx.x;
  if (t >= KSUB * NPTS) return;
  int q = t / NPTS, i = t - q * NPTS;
  V[t] = sinf(0.5f + 0.12345f * (float)i + 1.2345f * (float)q) +
         0.01f * (float)((i * (q + 3)) % 7);
}

// W[:,q] = K V[:,q] + shift*V[:,q]; one block per row i
__global__ __launch_bounds__(64) void ppgml_kv(const float* __restrict__ K,
                                               const float* __restrict__ Vin,
                                               float* __restrict__ Wout,
                                               const float* __restrict__ shiftSq) {
  __shared__ float red[KSUB][64];
  const int i = blockIdx.x;
  const float shift = sqrtf(fmaxf(*shiftSq, 0.0f));
  float acc[KSUB];
#pragma unroll
  for (int q = 0; q < KSUB; ++q) acc[q] = 0.0f;
  for (int j = threadIdx.x; j < NPTS; j += 64) {
    float kv = K[i * NPTS + j];
#pragma unroll
    for (int q = 0; q < KSUB; ++q) acc[q] += kv * Vin[q * NPTS + j];
  }
#pragma unroll
  for (int q = 0; q < KSUB; ++q) red[q][threadIdx.x] = acc[q];
  __syncthreads();
  for (int s = 32; s > 0; s >>= 1) {
    if (threadIdx.x < s)
#pragma unroll
      for (int q = 0; q < KSUB; ++q) red[q][threadIdx.x] += red[q][threadIdx.x + s];
    __syncthreads();
  }
  if (threadIdx.x < KSUB)
    Wout[threadIdx.x * NPTS + i] = red[threadIdx.x][0] + shift * Vin[threadIdx.x * NPTS + i];
}

__device__ inline float ppgml_red384(float v, float* red) {
  int i = threadIdx.x;
  red[i] = v; __syncthreads();
  for (int s = 256; s > 0; s >>= 1) {
    if (i < s && i + s < NPTS) red[i] += red[i + s];
    __syncthreads();
  }
  float r = red[0]; __syncthreads();
  return r;
}

// modified Gram-Schmidt, one block of 384 threads
__global__ __launch_bounds__(NPTS) void ppgml_ortho(float* __restrict__ W) {
  __shared__ float red[NPTS];
  __shared__ float coef;
  const int i = threadIdx.x;
  for (int q = 0; q < KSUB; ++q) {
    float* wq = W + q * NPTS;
    for (int p = 0; p < q; ++p) {
      const float* wp = W + p * NPTS;
      float dot = ppgml_red384(wq[i] * wp[i], red);
      if (i == 0) coef = dot;
      __syncthreads();
      wq[i] -= coef * wp[i];
      __syncthreads();
    }
    float nrm2 = ppgml_red384(wq[i] * wq[i], red);
    if (i == 0) coef = (nrm2 > 1e-30f) ? rsqrtf(nrm2) : 0.0f;
    __syncthreads();
    float v = wq[i] * coef;
    if (nrm2 <= 1e-30f) v = (i == q) ? 1.0f : 0.0f;  // degenerate fallback
    wq[i] = v;
    __syncthreads();
  }
}

__global__ void ppgml_copy(const float* __restrict__ src, float* __restrict__ dst, int n) {
  int t = blockIdx.x * blockDim.x + threadIdx.x;
  if (t < n) dst[t] = src[t];
}

// M = V^T (K V) -> 8x8
__global__ __launch_bounds__(64) void ppgml_rayleigh(const float* __restrict__ V,
                                                     const float* __restrict__ W,
                                                     float* __restrict__ M) {
  int e = threadIdx.x;
  if (e >= KSUB * KSUB) return;
  int p = e / KSUB, q = e - p * KSUB;
  float s = 0.0f;
  for (int j = 0; j < NPTS; ++j) s += V[p * NPTS + j] * W[q * NPTS + j];
  M[e] = s;
}

// 8x8 symmetric Jacobi; sorted descending eigenvalues + rotation U
__global__ void ppgml_jacobi8(const float* __restrict__ Min, float* __restrict__ Uout,
                              float* __restrict__ evals) {
  if (threadIdx.x != 0 || blockIdx.x != 0) return;
  float A[KSUB][KSUB], Q[KSUB][KSUB];
  for (int p = 0; p < KSUB; ++p)
    for (int q = 0; q < KSUB; ++q) { A[p][q] = Min[p * KSUB + q]; Q[p][q] = (p == q) ? 1.f : 0.f; }
  for (int p = 0; p < KSUB; ++p)
    for (int q = p + 1; q < KSUB; ++q) { float v = 0.5f * (A[p][q] + A[q][p]); A[p][q] = v; A[q][p] = v; }
  for (int sweep = 0; sweep < 30; ++sweep)
    for (int p = 0; p < KSUB; ++p)
      for (int q = p + 1; q < KSUB; ++q) {
        float apq = A[p][q];
        if (fabsf(apq) < 1e-12f) continue;
        float tau = (A[q][q] - A[p][p]) / (2.0f * apq);
        float tt  = ((tau >= 0.f) ? 1.f : -1.f) / (fabsf(tau) + sqrtf(1.f + tau * tau));
        float c   = rsqrtf(1.f + tt * tt), s = tt * c;
        for (int r = 0; r < KSUB; ++r) {
          float arp = A[r][p], arq = A[r][q];
          A[r][p] = c * arp - s * arq; A[r][q] = s * arp + c * arq;
        }
        for (int r = 0; r < KSUB; ++r) {
          float apr = A[p][r], aqr = A[q][r];
          A[p][r] = c * apr - s * aqr; A[q][r] = s * apr + c * aqr;
        }
        for (int r = 0; r < KSUB; ++r) {
          float qp = Q[r][p], qq = Q[r][q];
          Q[r][p] = c * qp - s * qq; Q[r][q] = s * qp + c * qq;
        }
      }
  int ord[KSUB];
  for (int a = 0; a < KSUB; ++a) ord[a] = a;
  for (int a = 0; a < KSUB; ++a)
    for (int b = a + 1; b < KSUB; ++b)
      if (A[ord[b]][ord[b]] > A[ord[a]][ord[a]]) { int t = ord[a]; ord[a] = ord[b]; ord[b] = t; }
  for (int c = 0; c < KSUB; ++c) {
    evals[c] = A[ord[c]][ord[c]];
    for (int q = 0; q < KSUB; ++q) Uout[q * KSUB + c] = Q[q][ord[c]];
  }
}

// Xp6[i][c] = (V U)[i][c] * sqrt(max(eval_c, 0)), c < 6
__global__ void ppgml_xp(const float* __restrict__ V, const float* __restrict__ U,
                         const float* __restrict__ evals, float* __restrict__ Xp) {
  int t = blockIdx.x * blockDim.x + threadIdx.x;
  if (t >= NPTS * 6) return;
  int i = t / 6, c = t - i * 6;
  float s = 0.0f;
#pragma unroll
  for (int q = 0; q < KSUB; ++q) s += V[q * NPTS + i] * U[q * KSUB + c];
  Xp[t] = s * sqrtf(fmaxf(evals[c], 0.0f));
}

// ---------------------------------------------------------------------------
// Per-dim small stages
// ---------------------------------------------------------------------------
__global__ void ppgml_pip(const float* __restrict__ Xp, const float* __restrict__ emb,
                          float* __restrict__ pips, int dd, int di) {
  __shared__ float tv[8];
  int c = threadIdx.x;
  if (c < dd) {
    float s = 0.0f;
    for (int i = 0; i < NPTS; ++i) s += Xp[i * 6 + c] * emb[i * DFEAT + (DFEAT - dd)];
    tv[c] = s;
  }
  __syncthreads();
  if (c == 0) {
    float p = 0.001f * (float)(DFEAT - dd);
    for (int q = 0; q < dd; ++q) p += 2.0f * tv[q] * tv[q];
    pips[di] = p;
  }
}

__global__ void ppgml_logmap(const float* __restrict__ Xp, float* __restrict__ L, int dd) {
  int i = blockIdx.x * blockDim.x + threadIdx.x;
  if (i >= NPTS) return;
  float x[6];
  float n2 = 0.0f;
  for (int c = 0; c < dd; ++c) { x[c] = Xp[i * 6 + c]; n2 += x[c] * x[c]; }
  float nrm = fmaxf(sqrtf(n2), 1e-12f);
  float mean = 0.0f;
  for (int c = 0; c < dd; ++c) { x[c] /= nrm; mean += x[c]; }
  mean /= (float)dd;
  float last  = fminf(fmaxf(x[dd - 1], -1.0f), 1.0f);
  float theta = acosf(last);
  float scale = theta / fmaxf(sinf(theta), 1e-20f);
  for (int c = 0; c < dd - 1; ++c) L[i * 5 + c] = (x[c] - mean) * scale;
}

__global__ __launch_bounds__(NPTS) void ppgml_ldastats(const float* __restrict__ L,
                                                       const int* __restrict__ labels,
                                                       float* __restrict__ means,
                                                       float* __restrict__ Lmean,
                                                       float* __restrict__ cnts, int k) {
  __shared__ float acc[9 * 5];
  __shared__ float cn[9];
  __shared__ float colsum[5];
  const int i = threadIdx.x;
  if (i < 45) acc[i] = 0.0f;
  if (i < 9)  cn[i] = 0.0f;
  if (i < 5)  colsum[i] = 0.0f;
  __syncthreads();
  int lab = labels[i];
  atomicAdd(&cn[lab], 1.0f);
  for (int c = 0; c < k; ++c) {
    float v = L[i * 5 + c];
    atomicAdd(&acc[lab * 5 + c], v);
    atomicAdd(&colsum[c], v);
  }
  __syncthreads();
  if (i < 45) means[i] = acc[i] / fmaxf(cn[i / 5], 1.0f);
  if (i < 9)  cnts[i] = cn[i];
  if (i < 5)  Lmean[i] = colsum[i] / (float)NPTS;
}

__global__ void ppgml_sw(const float* __restrict__ L, const int* __restrict__ labels,
                         const float* __restrict__ means, float* __restrict__ SW, int k) {
  int e = threadIdx.x;
  if (e >= k * k) return;
  int a = e / k, b = e - a * k;
  float s = 0.0f;
  for (int i = 0; i < NPTS; ++i) {
    int lab = labels[i];
    s += (L[i * 5 + a] - means[lab * 5 + a]) * (L[i * 5 + b] - means[lab * 5 + b]);
  }
  SW[e] = s;
}

__global__ void ppgml_sb(const float* __restrict__ means, const float* __restrict__ Lmean,
                         const float* __restrict__ cnts, float* __restrict__ SB, int k) {
  int e = threadIdx.x;
  if (e >= k * k) return;
  int a = e / k, b = e - a * k;
  float s = 0.0f;
  for (int c = 0; c < 9; ++c) {
    float va = means[c * 5 + a] - Lmean[a];
    float vb = means[c * 5 + b] - Lmean[b];
    float vd = (cnts[c] > 0.0f) ? 1.0f : 0.0f;
    s += vd * va * vb;
  }
  SB[e] = 4.0f * s;
}

// fisher = inv(SW + 1e-3 I) @ SB; eigmean = trace/k; topvec by power iteration
__global__ void ppgml_fisher(const float* __restrict__ SW, const float* __restrict__ SB,
                             float* __restrict__ topvec, float* __restrict__ eigm,
                             int k, int di) {
  if (threadIdx.x != 0 || blockIdx.x != 0) return;
  float A[5][10];
  for (int p = 0; p < k; ++p) {
    for (int q = 0; q < k; ++q) A[p][q] = SW[p * k + q] + ((p == q) ? 1e-3f : 0.0f);
    for (int q = 0; q < k; ++q) A[p][k + q] = (p == q) ? 1.0f : 0.0f;
  }
  for (int col = 0; col < k; ++col) {
    int piv = col;
    for (int r = col + 1; r < k; ++r)
      if (fabsf(A[r][col]) > fabsf(A[piv][col])) piv = r;
    if (piv != col)
      for (int c = 0; c < 2 * k; ++c) { float t = A[col][c]; A[col][c] = A[piv][c]; A[piv][c] = t; }
    float dv = A[col][col];
    if (fabsf(dv) < 1e-20f) dv = (dv >= 0.0f) ? 1e-20f : -1e-20f;
    float inv = 1.0f / dv;
    for (int c = 0; c < 2 * k; ++c) A[col][c] *= inv;
    for (int r = 0; r < k; ++r) {
      if (r == col) continue;
      float f = A[r][col];
      for (int c = 0; c < 2 * k; ++c) A[r][c] -= f * A[col][c];
    }
  }
  float F[5][5];
  for (int p = 0; p < k; ++p)
    for (int q = 0; q < k; ++q) {
      float s = 0.0f;
      for (int m = 0; m < k; ++m) s += A[p][k + m] * SB[m * k + q];
      F[p][q] = s;
    }
  float tr = 0.0f;
  for (int p = 0; p < k; ++p) tr += F[p][p];
  eigm[di] = tr / (float)k;
  float v[5];
  for (int p = 0; p < k; ++p) v[p] = 1.0f;
  for (int it = 0; it < 300; ++it) {
    float w[5], n2 = 0.0f;
    for (int p = 0; p < k; ++p) {
      float s = 0.0f;
      for (int q = 0; q < k; ++q) s += F[p][q] * v[q];
      w[p] = s; n2 += s * s;
    }
    if (n2 < 1e-30f) { for (int p = 0; p < k; ++p) v[p] = (p == 0) ? 1.0f : 0.0f; }
    else { float inv = rsqrtf(n2); for (int p = 0; p < k; ++p) v[p] = w[p] * inv; }
  }
  for (int c = 0; c < k; ++c) topvec[c] = v[c];
}

__global__ __launch_bounds__(NPTS) void ppgml_ref(const float* __restrict__ un,
                                                  const float* __restrict__ topvec,
                                                  const float* __restrict__ eigm,
                                                  const float* __restrict__ embsum,
                                                  float* __restrict__ refs, int k, int di) {
  __shared__ float red[NPTS];
  const int i = threadIdx.x;
  float center = fabsf(embsum[0] / (float)(NPTS * DFEAT)) + 0.5f;
  float ml = fabsf(eigm[di]) * 0.1f;
  float s = 0.0f;
  for (int c = 0; c < k; ++c)
    s += (center - ml + 2.0f * ml * un[i * 5 + c]) * topvec[c];
  float m = ppgml_red384(fabsf(s), red);
  if (i == 0) {
    float r = sqrtf(m / (float)NPTS);
    refs[di] = fminf(fmaxf(r, 0.7f), 1.5f);
  }
}

__global__ void ppgml_select(const float* __restrict__ pips, const float* __restrict__ refs,
                             const float* __restrict__ eigm, float* __restrict__ refsel,
                             float* __restrict__ eigsel) {
  if (threadIdx.x != 0 || blockIdx.x != 0) return;
  int best = 0;
  if (pips[1] < pips[best]) best = 1;
  if (pips[2] < pips[best]) best = 2;
  *refsel = refs[best];
  *eigsel = eigm[best];
}

// semihard triplet: one thread per (anchor, positive), loop over negatives
__global__ __launch_bounds__(256) void ppgml_triplet(const float* __restrict__ d,
                                                     const int* __restrict__ labels,
                                                     const float* __restrict__ refp,
                                                     float* __restrict__ sum,
                                                     float* __restrict__ cnt) {
  __shared__ float rs[256];
  __shared__ float rc[256];
  int t = blockIdx.x * blockDim.x + threadIdx.x;
  float lsum = 0.0f, lcnt = 0.0f;
  if (t < NN2) {
    int a = t / NPTS, p = t - a * NPTS;
    int la = labels[a];
    if (a != p && labels[p] == la) {
      const float ref = *refp;
      float dap = d[a * NPTS + p];
      dap = sqrtf(dap * dap + 1e-16f);
      for (int n = 0; n < NPTS; ++n) {
        if (labels[n] == la) continue;
        float dan = d[a * NPTS + n];
        dan = sqrtf(dan * dan + 1e-16f);
        float gap  = dan - dap;
        float viol = fmaxf(ref * dap - ref * dan + 0.1f, 0.0f);
        if (gap > 0.0f && gap < 0.5f && viol > 0.0f) { lsum += viol; lcnt += 1.0f; }
      }
    }
  }
  rs[threadIdx.x] = lsum; rc[threadIdx.x] = lcnt;
  __syncthreads();
  for (int s = 128; s > 0; s >>= 1) {
    if (threadIdx.x < s) { rs[threadIdx.x] += rs[threadIdx.x + s]; rc[threadIdx.x] += rc[threadIdx.x + s]; }
    __syncthreads();
  }
  if (threadIdx.x == 0) { atomicAdd(sum, rs[0]); atomicAdd(cnt, rc[0]); }
}

__global__ void ppgml_finalize(const float* __restrict__ sum, const float* __restrict__ cnt,
                               const float* __restrict__ eigsel, float* __restrict__ out) {
  if (threadIdx.x != 0 || blockIdx.x != 0) return;
  out[0] = sum[0] / fmaxf(cnt[0], 1.0f) - 1e-6f * eigsel[0];
}

__global__ void ppgml_zero(float* __restrict__ s, int n) {
  int t = blockIdx.x * blockDim.x + threadIdx.x;
  if (t < n) s[t] = 0.0f;
}

// ---------------------------------------------------------------------------
// Host-side orchestration (all on `stream`; graph-capture safe)
// ---------------------------------------------------------------------------
extern "C" void kernel_launch(void* const* d_in, const int* in_sizes, int n_in,
                              void* d_out, int out_size, void* d_ws, size_t ws_size,
                              hipStream_t stream) {
  (void)in_sizes; (void)n_in; (void)out_size; (void)ws_size;
  const float* emb    = (const float*)d_in[0];   // (384, 512) f32
  const float* unoise = (const float*)d_in[1];   // (384, 5)   f32
  const int*   labels = (const int*)d_in[2];     // (384,)     i32

  float* W0   = (float*)d_ws;
  float* gram = W0;                        // 147456
  float* dmat = W0 + 147456;               // 147456
  float* K    = W0 + 294912;               // 147456 (graph / FW / B / centered K)
  float* V    = W0 + 442368;               // 3072
  float* Wv   = W0 + 445440;               // 3072
  float* Xp   = W0 + 448512;               // 2304
  float* L    = W0 + 450816;               // 1920
  float* S    = W0 + 452736;               // 256 scalars
  float* rm   = W0 + 452992;               // 384
  int*   knn  = (int*)(W0 + 453376);       // 2304 ints

  // scalar slots
  float* sEmbSum = S + 0;
  float* sFmax   = S + 1;   // uint bits
  float* sTotB   = S + 2;
  float* sLSum   = S + 3;
  float* sLCnt   = S + 4;
  float* sRefSel = S + 5;
  float* sEigSel = S + 6;
  float* sKsq    = S + 7;
  float* sPips   = S + 8;   // [3]
  float* sRefs   = S + 11;  // [3]
  float* sEigm   = S + 14;  // [3]
  float* sZero   = S + 19;  // stays 0
  float* sEvals  = S + 24;  // [8]
  float* sM      = S + 32;  // [64]
  float* sU      = S + 96;  // [64]
  float* sMeans  = S + 160; // [45]
  float* sLmean  = S + 205; // [5]
  float* sCnts   = S + 210; // [9]
  float* sSW     = S + 220; // [25]
  float* sSB     = S + 245; // [25]
  float* sTopV   = S + 270; // [5]

  ppgml_zero<<<1, 256, 0, stream>>>(S, 256);
  ppgml_sumall<<<192, 256, 0, stream>>>(emb, NPTS * DFEAT, sEmbSum, 0);

  // Gram (WMMA) -> distances -> kNN graph
  ppgml_gram_wmma<<<576, 32, 0, stream>>>(emb, gram);
  ppgml_dist<<<576, 256, 0, stream>>>(gram, dmat);
  ppgml_knn<<<12, 32, 0, stream>>>(dmat, knn);
  ppgml_ginit<<<576, 256, 0, stream>>>(K);
  ppgml_gscatter<<<72, 32, 0, stream>>>(dmat, knn, K);
  ppgml_gsym<<<576, 256, 0, stream>>>(K);

  // geodesic distances + double centering
  ppgml_floyd<<<1, 1024, 0, stream>>>(K);
  ppgml_fmax<<<144, 256, 0, stream>>>(K, (unsigned*)sFmax);
  ppgml_bmat<<<NPTS, 128, 0, stream>>>(K, rm, sTotB, (const unsigned*)sFmax);
  ppgml_center<<<576, 256, 0, stream>>>(K, rm, sTotB);
  ppgml_sumall<<<144, 256, 0, stream>>>(K, NN2, sKsq, 1);  // ||K||_F^2 (shift)

  // shifted subspace iteration, then Rayleigh-Ritz
  ppgml_initv<<<12, 256, 0, stream>>>(V);
  for (int it = 0; it < 40; ++it) {
    ppgml_kv<<<NPTS, 64, 0, stream>>>(K, V, Wv, sKsq);
    ppgml_ortho<<<1, NPTS, 0, stream>>>(Wv);
    ppgml_copy<<<12, 256, 0, stream>>>(Wv, V, KSUB * NPTS);
  }
  ppgml_kv<<<NPTS, 64, 0, stream>>>(K, V, Wv, sZero);
  ppgml_rayleigh<<<1, 64, 0, stream>>>(V, Wv, sM);
  ppgml_jacobi8<<<1, 1, 0, stream>>>(sM, sU, sEvals);
  ppgml_xp<<<9, 256, 0, stream>>>(V, sU, sEvals, Xp);

  // per-dim: pip, logmap, LDA, ref
  for (int di = 0; di < 3; ++di) {
    const int dd = 4 + di;
    const int k  = dd - 1;
    ppgml_pip<<<1, 32, 0, stream>>>(Xp, emb, sPips, dd, di);
    ppgml_logmap<<<12, 32, 0, stream>>>(Xp, L, dd);
    ppgml_ldastats<<<1, NPTS, 0, stream>>>(L, labels, sMeans, sLmean, sCnts, k);
    ppgml_sw<<<1, 32, 0, stream>>>(L, labels, sMeans, sSW, k);
    ppgml_sb<<<1, 32, 0, stream>>>(sMeans, sLmean, sCnts, sSB, k);
    ppgml_fisher<<<1, 1, 0, stream>>>(sSW, sSB, sTopV, sEigm, k, di);
    ppgml_ref<<<1, NPTS, 0, stream>>>(unoise, sTopV, sEigm, sEmbSum, sRefs, k, di);
  }

  ppgml_select<<<1, 1, 0, stream>>>(sPips, sRefs, sEigm, sRefSel, sEigSel);
  ppgml_triplet<<<576, 256, 0, stream>>>(dmat, labels, sRefSel, sLSum, sLCnt);
  ppgml_finalize<<<1, 1, 0, stream>>>(sLSum, sLCnt, sEigSel, (float*)d_out);
}